// ConvertTRTFormatToFlatTensor_89601607729526
// MI455X (gfx1250) — compile-verified
//
#include <hip/hip_runtime.h>
#include <stdint.h>

#define TILE_ROWS 256
#define MAX_COLS 8

// ---------------------------------------------------------------------------
// Kernel 1: exclusive prefix sum of clamped num_predictions -> offs[0..B]
// Valid rows per batch are the prefix [0, n_b), so compaction offsets are a
// simple scan. One block, LDS Hillis-Steele scan.
// ---------------------------------------------------------------------------
__global__ void scan_offsets_kernel(const int* __restrict__ num_pred,
                                    int* __restrict__ offs, int B, int M) {
  __shared__ int s[1024];
  const int t = threadIdx.x;
  int v = 0;
  if (t < B) {
    v = num_pred[t];
    v = v < 0 ? 0 : (v > M ? M : v);
  }
  s[t] = v;
  __syncthreads();
  for (int d = 1; d < (int)blockDim.x; d <<= 1) {
    int x = (t >= d) ? s[t - d] : 0;
    __syncthreads();
    s[t] += x;
    __syncthreads();
  }
  if (t == 0) offs[0] = 0;
  if (t < B) offs[t + 1] = s[t];
}

// ---------------------------------------------------------------------------
// Kernel 2: gather-style compaction. One thread per output row; rows past the
// total valid count are zero. Rows are composed in an LDS tile and streamed
// out with the CDNA5 async LDS->global engine (coalesced b128 chunks).
// ---------------------------------------------------------------------------
__global__ void compact_rows_kernel(const float4* __restrict__ boxes,
                                    const float*  __restrict__ scores,
                                    const int*    __restrict__ classes,
                                    const int*    __restrict__ offs,
                                    float* __restrict__ out,
                                    int B, int M, int ncols,
                                    long long totalRows, long long outFloats) {
  __shared__ int s_offs[1025];
  __shared__ __align__(16) float s_tile[TILE_ROWS * MAX_COLS];

  for (int i = threadIdx.x; i <= B; i += blockDim.x) s_offs[i] = offs[i];
  __syncthreads();

  const long long r0 = (long long)blockIdx.x * TILE_ROWS;
  const long long r  = r0 + threadIdx.x;
  const int total    = s_offs[B];

  float row[MAX_COLS];
#pragma unroll
  for (int c = 0; c < MAX_COLS; ++c) row[c] = 0.0f;

  if (r < (long long)total) {
    // largest b with s_offs[b] <= r  (s_offs[0]=0 <= r < s_offs[B]=total)
    int lo = 0, hi = B;
    while (hi - lo > 1) {
      const int mid = (lo + hi) >> 1;
      if ((long long)s_offs[mid] <= r) lo = mid; else hi = mid;
    }
    const int b = lo;
    const int m = (int)(r - (long long)s_offs[b]);
    const long long idx = (long long)b * M + m;
    const float4 bx = boxes[idx];          // global_load_b128, 16B aligned
    row[0] = (float)b;
    row[1] = bx.x; row[2] = bx.y; row[3] = bx.z; row[4] = bx.w;
    row[5] = scores[idx];
    row[6] = (float)classes[idx];
  }

  if (r < totalRows) {
    // stride ncols (7) words is coprime with 64 banks -> conflict-free
    float* dstrow = &s_tile[(int)threadIdx.x * ncols];
    for (int c = 0; c < ncols; ++c) dstrow[c] = row[c];
  }
  __syncthreads();

  // Cooperative store of this block's contiguous output region.
  const long long baseFloat = r0 * (long long)ncols;
  long long validFloats = (totalRows - r0) * (long long)ncols;
  const long long tileFloats = (long long)TILE_ROWS * ncols;
  if (validFloats > tileFloats) validFloats = tileFloats;
  if (baseFloat + validFloats > outFloats) validFloats = outFloats - baseFloat;
  if (validFloats < 0) validFloats = 0;
  const int validBytes = (int)(validFloats * 4);
  const int vecBytes   = validBytes & ~15;   // 16B-granular portion

#if __has_builtin(__builtin_amdgcn_global_store_async_from_lds_b128)
  {
    typedef int v4i_t __attribute__((vector_size(16)));
    typedef __attribute__((address_space(1))) char  gchar_t;
    typedef __attribute__((address_space(3))) char  lchar_t;
    typedef __attribute__((address_space(1))) v4i_t gvec_t;
    typedef __attribute__((address_space(3))) v4i_t lvec_t;
    // Global flat address == AS1 address; LDS flat address low 32 bits ==
    // LDS offset (shared aperture lives in the high 32 bits).
    gchar_t* gbase = (gchar_t*)(uintptr_t)((char*)out + baseFloat * 4);
    lchar_t* lbase = (lchar_t*)(uint32_t)(uintptr_t)((char*)s_tile);
    for (int off = (int)threadIdx.x * 16; off < vecBytes;
         off += (int)blockDim.x * 16) {
      __builtin_amdgcn_global_store_async_from_lds_b128(
          (gvec_t*)(gbase + off), (lvec_t*)(lbase + off), 0, 0);
    }
    asm volatile("s_wait_asynccnt 0" ::: "memory");
  }
#else
  {
    float4* gdst = (float4*)((char*)out + baseFloat * 4);
    const float4* lsrc = (const float4*)s_tile;
    for (int off = (int)threadIdx.x; off * 16 < vecBytes;
         off += (int)blockDim.x) {
      gdst[off] = lsrc[off];
    }
  }
#endif

  // scalar tail (only if region not a multiple of 16B; never with these sizes)
  for (int f = (vecBytes >> 2) + (int)threadIdx.x; f < (validBytes >> 2);
       f += (int)blockDim.x) {
    out[baseFloat + f] = s_tile[f];
  }
}

// ---------------------------------------------------------------------------
// Host launcher
// inputs (setup_inputs order): num_predictions[B], pred_boxes[B*M*4],
//                              pred_scores[B*M], pred_classes[B*M]
// ---------------------------------------------------------------------------
extern "C" void kernel_launch(void* const* d_in, const int* in_sizes, int n_in,
                              void* d_out, int out_size, void* d_ws,
                              size_t ws_size, hipStream_t stream) {
  (void)n_in; (void)ws_size;
  const int*    num_pred = (const int*)d_in[0];
  const float4* boxes    = (const float4*)d_in[1];
  const float*  scores   = (const float*)d_in[2];
  const int*    classes  = (const int*)d_in[3];
  float*        out      = (float*)d_out;

  const int B = in_sizes[0];                 // num_predictions is [B,1]
  const int M = in_sizes[2] / B;             // pred_scores is [B,M]
  const long long totalRows = (long long)B * M;
  int ncols = (int)((long long)out_size / totalRows);
  if (ncols < 1) ncols = 1;
  if (ncols > MAX_COLS) ncols = MAX_COLS;

  int* offs = (int*)d_ws;                    // B+1 ints of scratch

  int scanThreads = 1;
  while (scanThreads < B) scanThreads <<= 1; // pow2 >= B (B=128 -> 128)
  if (scanThreads > 1024) scanThreads = 1024;
  scan_offsets_kernel<<<1, scanThreads, 0, stream>>>(num_pred, offs, B, M);

  const long long nBlocks = (totalRows + TILE_ROWS - 1) / TILE_ROWS;
  compact_rows_kernel<<<dim3((unsigned)nBlocks), dim3(TILE_ROWS), 0, stream>>>(
      boxes, scores, classes, offs, out, B, M, ncols, totalRows,
      (long long)out_size);
}